// Loss_yolov1_55224689492071
// MI455X (gfx1250) — compile-verified
//
#include <hip/hip_runtime.h>
#include <hip/hip_bf16.h>
#include <stdint.h>

// Problem geometry (fixed by the reference)
#define NB     16384
#define NCH    30
#define SS     49                 // 7*7 cells
#define IMG_F  (NCH * SS)         // 1470 floats per image per tensor
#define BT     4                  // images per tile (even -> 16B-aligned tiles)
#define TILE_F (BT * IMG_F)       // 5880 floats per tensor per tile
#define TILE_BYTES (TILE_F * 4)   // 23520 bytes (multiple of 16)
#define BLOCK  256
#define TCHUNK (TILE_BYTES / 16)  // 1470 B128 chunks per tensor per tile
#define FULLR  (TCHUNK / BLOCK)   // 5 full copy rounds per tensor
#define OPS_PER_TENSOR (FULLR + 1)        // 6 async ops per thread per tensor
#define OPS_PER_TILE   (2 * OPS_PER_TENSOR) // 12 async ops per thread per tile
#define NWG    512
#define TILES  (NB / BT)          // 4096
#define TPW    (TILES / NWG)      // 8 tiles per workgroup
#define BUF_F  (2 * TILE_F)       // 11760 floats per LDS buffer (pred then labels)

static_assert(NWG * BT * TPW == NB, "grid must cover batch exactly");
static_assert(TILE_BYTES % 16 == 0, "tile must be 16B aligned");

#if __has_builtin(__builtin_amdgcn_global_load_async_to_lds_b128)
#define HAVE_ASYNC 1
#else
#define HAVE_ASYNC 0
#endif

#if __has_builtin(__builtin_amdgcn_s_wait_asynccnt)
#define WAIT_ASYNC(n) __builtin_amdgcn_s_wait_asynccnt(n)
#else
#define WAIT_ASYNC(n) asm volatile("s_wait_asynccnt %0" ::"n"(n) : "memory")
#endif

typedef int v4i __attribute__((ext_vector_type(4)));
typedef __attribute__((address_space(1))) v4i* gv4i_t;  // global (device) int4*
typedef __attribute__((address_space(3))) v4i* lv4i_t;  // LDS (shared) int4*

__device__ __forceinline__ void copy16(const char* src, char* dst) {
#if HAVE_ASYNC
  __builtin_amdgcn_global_load_async_to_lds_b128(
      (gv4i_t)const_cast<char*>(src), (lv4i_t)dst, 0, 0);
#else
  *(float4*)dst = *(const float4*)src;
#endif
}

// Stage one 23520-byte tensor tile into LDS: 6 branchless B128 ops per thread.
__device__ __forceinline__ void issue_tensor(const char* gbase, char* lbase, int tid) {
#pragma unroll
  for (int r = 0; r < FULLR; ++r) {
    const size_t off = (size_t)(tid + r * BLOCK) * 16;  // affine: tid*16 + r*4096
    copy16(gbase + off, lbase + off);
  }
  // tail round: chunks [1280,1470); clamp (duplicates are benign, keeps op count uniform)
  unsigned c = (unsigned)(tid + FULLR * BLOCK);
  c = (c < TCHUNK) ? c : (TCHUNK - 1u);
  const size_t off = (size_t)c * 16;
  copy16(gbase + off, lbase + off);
}

__device__ __forceinline__ void issue_tile(const float* __restrict__ pred,
                                           const float* __restrict__ labels,
                                           int tile, float* smem_buf, int tid) {
  const char* gp = (const char*)pred + (size_t)tile * TILE_BYTES;
  const char* gl = (const char*)labels + (size_t)tile * TILE_BYTES;
  char* lb = (char*)smem_buf;
  issue_tensor(gp, lb, tid);
  issue_tensor(gl, lb + TILE_BYTES, tid);
}

__device__ __forceinline__ float iou_ref(float x1a, float y1a, float x2a, float y2a,
                                         float x1b, float y1b, float x2b, float y2b) {
  bool no_overlap = (x2a < x1b) || (x1a > x2b) || (y2a < y1b) || (y1a > y2b);
  float iw = fminf(x2a, x2b) - fmaxf(x1a, x1b);
  float ih = fminf(y2a, y2b) - fmaxf(y1a, y1b);
  float inter = no_overlap ? 0.0f : iw * ih;
  float area_a = (x2a - x1a) * (y2a - y1a);
  float area_b = (x2b - x1b) * (y2b - y1b);
  float uni = area_a + area_b - inter;
  return (inter > 0.0f) ? (inter / uni) : 0.0f;
}

__device__ __forceinline__ void to_xyxy(float x, float y, float w, float h,
                                        float fi, float fj,
                                        float& x1, float& y1, float& x2, float& y2) {
  float cx = (x + fi) / 7.0f;
  float cy = (y + fj) / 7.0f;
  x1 = cx - 0.5f * w;
  y1 = cy - 0.5f * h;
  x2 = cx + 0.5f * w;
  y2 = cy + 0.5f * h;
}

// P/L point at cell 'rem' of image 'img' inside LDS; channel c lives at [c*SS].
__device__ __forceinline__ float cell_loss(const float* P, const float* L,
                                           float fi, float fj) {
  float p0 = P[0], p1 = P[SS], p2 = P[2 * SS], p3 = P[3 * SS], p5 = P[5 * SS];
  float p6 = P[6 * SS], p7 = P[7 * SS], p8 = P[8 * SS], p9 = P[9 * SS], p11 = P[11 * SS];
  float l0 = L[0], l1 = L[SS], l2 = L[2 * SS], l3 = L[3 * SS], l4 = L[4 * SS];
  float l6 = L[6 * SS], l7 = L[7 * SS], l8 = L[8 * SS], l9 = L[9 * SS];

  float ax1, ay1, ax2, ay2, bx1, by1, bx2, by2, gx1, gy1, gx2, gy2;
  to_xyxy(p0, p1, p2, p3, fi, fj, ax1, ay1, ax2, ay2);
  to_xyxy(p6, p7, p8, p9, fi, fj, bx1, by1, bx2, by2);
  to_xyxy(l0, l1, l2, l3, fi, fj, gx1, gy1, gx2, gy2);

  float iou1 = iou_ref(ax1, ay1, ax2, ay2, gx1, gy1, gx2, gy2);
  float iou2 = iou_ref(bx1, by1, bx2, by2, gx1, gy1, gx2, gy2);
  bool use1 = iou1 >= iou2;

  float e0 = p0 - l0, e1 = p1 - l1;
  float s2 = sqrtf(p2) - sqrtf(l2), s3 = sqrtf(p3) - sqrtf(l3);
  float c1 = 5.0f * (e0 * e0 + e1 * e1 + s2 * s2 + s3 * s3);
  float e6 = p6 - l6, e7 = p7 - l7;
  float s8 = sqrtf(p8) - sqrtf(l8), s9 = sqrtf(p9) - sqrtf(l9);
  float c2 = 5.0f * (e6 * e6 + e7 * e7 + s8 * s8 + s9 * s9);
  float coor = use1 ? c1 : c2;

  float d1 = (p5 - iou1) * (p5 - iou1);
  float d2 = (p11 - iou2) * (p11 - iou2);
  float obj_conf = use1 ? d1 : d2;
  float noobj_in_obj = 0.5f * (use1 ? d2 : d1);

  float cls = 0.0f;
#pragma unroll
  for (int c = 12; c < 30; ++c) {
    float d = P[c * SS] - L[c * SS];
    cls = fmaf(d, d, cls);
  }

  float per_obj = coor + obj_conf + noobj_in_obj + cls;
  float per_noobj = 0.5f * (p5 * p5 + p11 * p11);
  return (l4 == 1.0f) ? per_obj : per_noobj;
}

__global__ __launch_bounds__(BLOCK) void yolo_loss_main(
    const float* __restrict__ pred, const float* __restrict__ labels,
    float* __restrict__ part) {
  __shared__ __align__(16) float smem[2 * BUF_F];  // 94080 B double buffer
  __shared__ float red[BLOCK / 32];

  const int tid = threadIdx.x;
  const int tile0 = blockIdx.x * TPW;
  float acc = 0.0f;

  issue_tile(pred, labels, tile0, smem, tid);

  for (int k = 0; k < TPW; ++k) {
    const int buf = k & 1;
    if (k + 1 < TPW) issue_tile(pred, labels, tile0 + k + 1, smem + (buf ^ 1) * BUF_F, tid);
#if HAVE_ASYNC
    if (k + 1 < TPW) {
      WAIT_ASYNC(OPS_PER_TILE);  // tile k done once only tile k+1's 12 ops remain
    } else {
      WAIT_ASYNC(0);
    }
#endif
    __syncthreads();

    if (tid < BT * SS) {
      const int img = tid / SS;
      const int rem = tid % SS;
      const float fi = (float)(rem / 7);  // reference: m broadcasts over dim -2
      const float fj = (float)(rem % 7);
      const float* P = smem + buf * BUF_F + img * IMG_F + rem;
      const float* L = P + TILE_F;
      acc += cell_loss(P, L, fi, fj);
    }
    __syncthreads();  // protect buf before it is overwritten by next issue
  }

  // deterministic in-block reduction
  float v = acc;
#pragma unroll
  for (int off = 16; off > 0; off >>= 1) v += __shfl_xor(v, off, 32);
  if ((tid & 31) == 0) red[tid >> 5] = v;
  __syncthreads();
  if (tid == 0) {
    float s = 0.0f;
#pragma unroll
    for (int w = 0; w < BLOCK / 32; ++w) s += red[w];
    part[blockIdx.x] = s;
  }
}

__global__ __launch_bounds__(NWG) void yolo_loss_reduce(
    const float* __restrict__ part, float* __restrict__ out) {
  __shared__ float red[NWG / 32];
  const int tid = threadIdx.x;
  float v = part[tid];
#pragma unroll
  for (int off = 16; off > 0; off >>= 1) v += __shfl_xor(v, off, 32);
  if ((tid & 31) == 0) red[tid >> 5] = v;
  __syncthreads();
  if (tid == 0) {
    float s = 0.0f;
#pragma unroll
    for (int w = 0; w < NWG / 32; ++w) s += red[w];
    out[0] = s * (1.0f / (float)NB);
  }
}

extern "C" void kernel_launch(void* const* d_in, const int* in_sizes, int n_in,
                              void* d_out, int out_size, void* d_ws, size_t ws_size,
                              hipStream_t stream) {
  const float* pred = (const float*)d_in[0];
  const float* labels = (const float*)d_in[1];
  float* out = (float*)d_out;
  float* part = (float*)d_ws;  // NWG floats of scratch

  yolo_loss_main<<<NWG, BLOCK, 0, stream>>>(pred, labels, part);
  yolo_loss_reduce<<<1, NWG, 0, stream>>>(part, out);
}